// nnComb_15281493639793
// MI455X (gfx1250) — compile-verified
//
#include <hip/hip_runtime.h>
#include <hip/hip_bf16.h>
#include <math.h>

typedef float v2f __attribute__((ext_vector_type(2)));
typedef float v4f __attribute__((ext_vector_type(4)));
typedef float v8f __attribute__((ext_vector_type(8)));

#define ROWS 16384   // B*CH*CH

// ---------------------------------------------------------------------------
// GEMM: C[M,N] = act(A)[M,K] @ W[K,N] + bias, fp32 WMMA 16x16x4.
// MODE 0: A is row-major [M,K]  (staged via async global->LDS B128)
// MODE 1: A[m,k] = |data0[b, k, rem]|, m = b*256 + rem  (data0: (64,1025,256))
// Block tile 64x64, K-chunk 16, 256 threads = 8 waves, 2 C-tiles per wave.
// Main K loop is branch-free; the lone ragged chunk (layer0 K=1025) is a
// single guarded epilogue iteration.
// ---------------------------------------------------------------------------
template <int MODE>
__global__ __launch_bounds__(256)
void gemm_wmma_f32(const float* __restrict__ A, const float* __restrict__ data0,
                   const float* __restrict__ W, const float* __restrict__ bias,
                   float* __restrict__ C, int N, int K)
{
  __shared__ float As[64][20];   // row stride 80B: 16B aligned, conflict-free
  __shared__ float Ws[16][68];   // row stride 272B: 16B aligned, conflict-free
  const int t    = threadIdx.x;
  const int lane = t & 31;
  const int wave = t >> 5;
  const int l15  = lane & 15;
  const int kh   = (lane >> 4) << 1;     // 0 for lanes 0-15, 2 for lanes 16-31
  const int row0 = blockIdx.y * 64;
  const int col0 = blockIdx.x * 64;
  const int tr   = wave & 3;             // tile row 0..3
  const int tc0  = (wave >> 2) << 1;     // tile cols {0,1} or {2,3}
  const int tc1  = tc0 + 1;

  v8f acc0 = {};
  v8f acc1 = {};

  // loader coordinates
  const int wkk = t >> 4, wnn = (t & 15) << 2;          // W tile: 16x64
  const int ar  = t >> 2, akb = (t & 3) << 2;           // A tile (mode 0): 64x16
  const int akk = t >> 4, arr = (t & 15) << 2;          // A tile (mode 1): 64x16
  const unsigned wlds = (unsigned)(uintptr_t)&Ws[wkk][wnn];
  const unsigned alds = (unsigned)(uintptr_t)&As[ar][akb];

  const int arow = tr * 16 + l15;
  const int bc0  = tc0 * 16 + l15;
  const int bc1  = tc1 * 16 + l15;

  auto mma16 = [&]() {
    #pragma unroll
    for (int kk = 0; kk < 16; kk += 4) {
      v2f af;  af.x  = As[arow][kk + kh];  af.y  = As[arow][kk + kh + 1];
      v2f b0v; b0v.x = Ws[kk + kh][bc0];   b0v.y = Ws[kk + kh + 1][bc0];
      acc0 = __builtin_amdgcn_wmma_f32_16x16x4_f32(false, af, false, b0v,
                                                   (short)0, acc0, false, false);
      v2f b1v; b1v.x = Ws[kk + kh][bc1];   b1v.y = Ws[kk + kh + 1][bc1];
      acc1 = __builtin_amdgcn_wmma_f32_16x16x4_f32(false, af, false, b1v,
                                                   (short)0, acc1, false, false);
    }
  };

  const int kmain = K & ~15;
  for (int k0 = 0; k0 < kmain; k0 += 16) {
    // ---- W tile: pure copy -> async data mover, no VGPR round-trip ----
    {
      const float* src = W + (size_t)(k0 + wkk) * N + col0 + wnn;
      asm volatile("global_load_async_to_lds_b128 %0, %1, off"
                   :: "v"(wlds), "v"(src) : "memory");
    }
    // ---- A tile ----
    if (MODE == 0) {
      const float* src = A + (size_t)(row0 + ar) * K + k0 + akb;
      asm volatile("global_load_async_to_lds_b128 %0, %1, off"
                   :: "v"(alds), "v"(src) : "memory");
    } else {
      // abs(data0) transposed: a (b,f) slab is 256 contiguous rows -> coalesced
      const v4f vv = *(const v4f*)(data0 +
          ((size_t)(row0 >> 8) * 1025 + k0 + akk) * 256 + (row0 & 255) + arr);
      As[arr + 0][akk] = fabsf(vv.x);
      As[arr + 1][akk] = fabsf(vv.y);
      As[arr + 2][akk] = fabsf(vv.z);
      As[arr + 3][akk] = fabsf(vv.w);
    }
    asm volatile("s_wait_asynccnt 0" ::: "memory");
    __syncthreads();
    mma16();
    __syncthreads();
  }

  if (kmain < K) {          // ragged final chunk (layer 0 only: k = 1024)
    {
      int k = kmain + wkk;
      if (k < K) {
        *(v4f*)&Ws[wkk][wnn] = *(const v4f*)(W + (size_t)k * N + col0 + wnn);
      } else {
        v4f zz = {};
        *(v4f*)&Ws[wkk][wnn] = zz;
      }
    }
    if (MODE == 0) {
      #pragma unroll
      for (int i = 0; i < 4; ++i) {
        int kk = akb + i;
        As[ar][kk] = (kmain + kk < K)
                   ? A[(size_t)(row0 + ar) * K + kmain + kk] : 0.f;
      }
    } else {
      int k = kmain + akk;
      if (k < K) {
        const v4f vv = *(const v4f*)(data0 +
            ((size_t)(row0 >> 8) * 1025 + k) * 256 + (row0 & 255) + arr);
        As[arr + 0][akk] = fabsf(vv.x);
        As[arr + 1][akk] = fabsf(vv.y);
        As[arr + 2][akk] = fabsf(vv.z);
        As[arr + 3][akk] = fabsf(vv.w);
      } else {
        As[arr + 0][akk] = 0.f; As[arr + 1][akk] = 0.f;
        As[arr + 2][akk] = 0.f; As[arr + 3][akk] = 0.f;
      }
    }
    __syncthreads();
    mma16();
  }

  // C/D layout: lane<16 -> M = tileRow + v, lane>=16 -> M = tileRow + 8 + v
  int crow = row0 + tr * 16 + ((lane >> 4) << 3);
  int c0 = col0 + tc0 * 16 + l15;
  int c1 = col0 + tc1 * 16 + l15;
  float bv0 = bias[c0], bv1 = bias[c1];
  #pragma unroll
  for (int v = 0; v < 8; ++v) {
    C[(size_t)(crow + v) * N + c0] = acc0[v] + bv0;
    C[(size_t)(crow + v) * N + c1] = acc1[v] + bv1;
  }
}

// ---------------------------------------------------------------------------
// LayerNorm(+ReLU) over last dim, one wave32 per row, shuffle reductions.
// ---------------------------------------------------------------------------
template <int N>
__global__ __launch_bounds__(256)
void ln_relu(float* __restrict__ X, const float* __restrict__ g,
             const float* __restrict__ e)
{
  const int lane = threadIdx.x & 31;
  const int row  = blockIdx.x * 8 + (threadIdx.x >> 5);
  constexpr int NP = N / 32;
  float xv[NP];
  float* p = X + (size_t)row * N;
  float s = 0.f;
  #pragma unroll
  for (int i = 0; i < NP; ++i) { xv[i] = p[lane + (i << 5)]; s += xv[i]; }
  #pragma unroll
  for (int off = 16; off; off >>= 1) s += __shfl_xor(s, off, 32);
  float mean = s * (1.f / N);
  float v = 0.f;
  #pragma unroll
  for (int i = 0; i < NP; ++i) { float d = xv[i] - mean; v += d * d; }
  #pragma unroll
  for (int off = 16; off; off >>= 1) v += __shfl_xor(v, off, 32);
  float rs = rsqrtf(v * (1.f / N) + 1e-5f);
  #pragma unroll
  for (int i = 0; i < NP; ++i) {
    int col = lane + (i << 5);
    float val = (xv[i] - mean) * rs * g[col] + e[col];
    p[col] = fmaxf(val, 0.f);
  }
}

// ---------------------------------------------------------------------------
// Head: p[r,0:4] = x[r,0:64] @ wf(64,4) + bf. One thread per row.
// ---------------------------------------------------------------------------
__global__ void head_gemv(const float* __restrict__ X, const float* __restrict__ wf,
                          const float* __restrict__ bf, float* __restrict__ p)
{
  int r = blockIdx.x * blockDim.x + threadIdx.x;
  if (r >= ROWS) return;
  const float* x = X + (size_t)r * 64;
  float a0 = bf[0], a1 = bf[1], a2 = bf[2], a3 = bf[3];
  #pragma unroll 4
  for (int k = 0; k < 64; ++k) {
    float xv = x[k];
    a0 += xv * wf[k * 4 + 0]; a1 += xv * wf[k * 4 + 1];
    a2 += xv * wf[k * 4 + 2]; a3 += xv * wf[k * 4 + 3];
  }
  float* o = p + (size_t)r * 4;
  o[0] = a0; o[1] = a1; o[2] = a2; o[3] = a3;
}

// ---------------------------------------------------------------------------
// Biquad coefficients per (b, section, I, J) -- frequency-independent, hoisted.
// torch-.view scramble: idx = s*512+par*256+I*16+J <-> i=idx>>6, j=(idx>>2)&15,
// q=idx&3 of the (B,16,16,4) head output.
// dB clip collapses to clamp(sigmoid, 1e-3, 10^-0.005).
// ---------------------------------------------------------------------------
__global__ void biquad_coef(const float* __restrict__ p, float* __restrict__ coef)
{
  int t = blockIdx.x * blockDim.x + threadIdx.x;   // 64*2*256 = 32768
  if (t >= 32768) return;
  int b = t >> 9;
  int s = (t >> 8) & 1;
  int I = (t >> 4) & 15;
  int J = t & 15;
  float pv[2];
  #pragma unroll
  for (int par = 0; par < 2; ++par) {
    int i = (s << 3) | (par << 2) | (I >> 2);
    int j = ((I & 3) << 2) | (J >> 2);
    int q = J & 3;
    pv[par] = p[((size_t)(b * 256 + i * 16 + j)) * 4 + q];
  }
  float fc = 1.f / (1.f + __expf(-0.25f * pv[0]));
  float gl = 1.f / (1.f + __expf(-0.25f * pv[1]));
  gl = fminf(fmaxf(gl, 1.0e-3f), 0.98855309f);     // clip(-60,-0.1) dB, exponentiated
  float sw, cw;
  __sincosf(3.14159265358979f * fc, &sw, &cw);
  float alpha = sw * 0.70710678f;                   // sin/(2Q), Q=1/sqrt(2)
  float b0 = gl * (1.f - cw) * 0.5f;
  float* c = coef + (size_t)t * 6;
  c[0] = b0; c[1] = b0 + b0; c[2] = b0;
  c[3] = 1.f + alpha; c[4] = -2.f * cw; c[5] = 1.f - alpha;
}

// ---------------------------------------------------------------------------
// Direct rDFT of z (2048,16), plus Dph and rhs = Dph .* X per (f, ch).
// Exact phase via (k*n mod 2048). One block per frequency bin.
// ---------------------------------------------------------------------------
__global__ __launch_bounds__(256)
void rfft_rhs(const float* __restrict__ z, const int* __restrict__ delays,
              float* __restrict__ Xre, float* __restrict__ Xim,
              float* __restrict__ rre, float* __restrict__ rim,
              float* __restrict__ dre_, float* __restrict__ dim_)
{
  __shared__ float lre[256], lim[256];
  const int k = blockIdx.x;
  const int t = threadIdx.x;
  const int c = t & 15, part = t >> 4;
  const float w0 = -6.28318530717959f / 2048.f;
  float sr = 0.f, si = 0.f;
  int nbeg = part * 128;
  for (int n = nbeg; n < nbeg + 128; ++n) {
    int m = (k * n) & 2047;
    float sn, cs;
    __sincosf(w0 * (float)m, &sn, &cs);
    float zv = z[n * 16 + c];
    sr += zv * cs; si += zv * sn;
  }
  lre[t] = sr; lim[t] = si;
  __syncthreads();
  if (t < 16) {
    float ar = 0.f, ai = 0.f;
    for (int pt = 0; pt < 16; ++pt) { ar += lre[pt * 16 + t]; ai += lim[pt * 16 + t]; }
    int o = k * 16 + t;
    Xre[o] = ar; Xim[o] = ai;
    int md = (k * delays[t]) & 2047;
    float dsn, dcs;
    __sincosf(w0 * (float)md, &dsn, &dcs);
    dre_[o] = dcs; dim_[o] = dsn;
    rre[o] = dcs * ar - dsn * ai;                   // (Dmat @ X)_i = Dph_i * X_i
    rim[o] = dcs * ai + dsn * ar;
  }
}

// ---------------------------------------------------------------------------
// Per (b,f): build A = I - diag(Dph)*Hb (Hb = 2-section RBJ cascade at z1),
// solve A y = Dph.*X (single RHS!) with partial-pivot Gaussian elimination
// in LDS, write |y|. One 256-thread block per system; thread (I,J) owns A[I][J].
// ---------------------------------------------------------------------------
__global__ __launch_bounds__(256)
void biquad_solve(const float* __restrict__ coef,
                  const float* __restrict__ rre, const float* __restrict__ rim,
                  const float* __restrict__ dre_, const float* __restrict__ dim_,
                  float* __restrict__ out)
{
  __shared__ float Are[16][17], Aim[16][17];
  __shared__ float yre[16], yim[16], mre[16], mim[16];
  __shared__ float pvr, pvi;
  __shared__ int prow;
  const int f = blockIdx.x, b = blockIdx.y;
  const int t = threadIdx.x;
  const int I = t >> 4, J = t & 15;

  // z1 = exp(-2*pi*i*f/2048), z2 = z1^2
  float z1i, z1r;
  __sincosf((-6.28318530717959f / 2048.f) * (float)f, &z1i, &z1r);
  float z2r = z1r * z1r - z1i * z1i;
  float z2i = 2.f * z1r * z1i;

  float hr = 1.f, hi = 0.f;                         // cascade product over sections
  #pragma unroll
  for (int s = 0; s < 2; ++s) {
    const float* cf = coef + ((size_t)((b * 2 + s) * 256 + t)) * 6;
    float c0 = cf[0], c1 = cf[1], c2 = cf[2], a0 = cf[3], a1 = cf[4], a2 = cf[5];
    float nr = c0 + c1 * z1r + c2 * z2r;
    float ni =      c1 * z1i + c2 * z2i;
    float dr = a0 + a1 * z1r + a2 * z2r;
    float di =      a1 * z1i + a2 * z2i;
    float inv = 1.f / (dr * dr + di * di);
    float sr2 = (nr * dr + ni * di) * inv;
    float si2 = (ni * dr - nr * di) * inv;
    float tr2 = hr * sr2 - hi * si2;
    hi = hr * si2 + hi * sr2;
    hr = tr2;
  }
  float dr = dre_[f * 16 + I], di = dim_[f * 16 + I];
  Are[I][J] = ((I == J) ? 1.f : 0.f) - (dr * hr - di * hi);
  Aim[I][J] = -(dr * hi + di * hr);
  if (t < 16) { yre[t] = rre[f * 16 + t]; yim[t] = rim[f * 16 + t]; }
  __syncthreads();

  for (int k = 0; k < 16; ++k) {
    if (t == 0) {
      int pp = k; float best = fabsf(Are[k][k]) + fabsf(Aim[k][k]);
      for (int i2 = k + 1; i2 < 16; ++i2) {
        float mg = fabsf(Are[i2][k]) + fabsf(Aim[i2][k]);
        if (mg > best) { best = mg; pp = i2; }
      }
      prow = pp;
    }
    __syncthreads();
    int pp = prow;
    if (pp != k) {
      if (t < 16) {
        float tmp = Are[k][t]; Are[k][t] = Are[pp][t]; Are[pp][t] = tmp;
        tmp = Aim[k][t]; Aim[k][t] = Aim[pp][t]; Aim[pp][t] = tmp;
      } else if (t == 16) {
        float tmp = yre[k]; yre[k] = yre[pp]; yre[pp] = tmp;
        tmp = yim[k]; yim[k] = yim[pp]; yim[pp] = tmp;
      }
    }
    __syncthreads();
    if (t == 0) {
      float d0 = Are[k][k], e0 = Aim[k][k];
      float inv = 1.f / (d0 * d0 + e0 * e0 + 1e-30f);
      pvr = d0 * inv; pvi = -e0 * inv;
    }
    __syncthreads();
    if (I > k && J == k) {
      float ar = Are[I][k], ai = Aim[I][k];
      mre[I] = ar * pvr - ai * pvi;
      mim[I] = ar * pvi + ai * pvr;
    }
    __syncthreads();
    if (I > k) {
      if (J > k) {
        float kr = Are[k][J], ki = Aim[k][J];
        Are[I][J] -= mre[I] * kr - mim[I] * ki;
        Aim[I][J] -= mre[I] * ki + mim[I] * kr;
      } else if (J == 0) {
        float ykr = yre[k], yki = yim[k];
        yre[I] -= mre[I] * ykr - mim[I] * yki;
        yim[I] -= mre[I] * yki + mim[I] * ykr;
      }
    }
    __syncthreads();
  }
  if (t == 0) {                                     // back-substitution (serial;
    for (int k = 15; k >= 0; --k) {                 // hidden by 65600 blocks)
      float sr2 = yre[k], si2 = yim[k];
      for (int j = k + 1; j < 16; ++j) {
        sr2 -= Are[k][j] * yre[j] - Aim[k][j] * yim[j];
        si2 -= Are[k][j] * yim[j] + Aim[k][j] * yre[j];
      }
      float d0 = Are[k][k], e0 = Aim[k][k];
      float inv = 1.f / (d0 * d0 + e0 * e0 + 1e-30f);
      yre[k] = (sr2 * d0 + si2 * e0) * inv;
      yim[k] = (si2 * d0 - sr2 * e0) * inv;
    }
  }
  __syncthreads();
  if (t < 16) {
    float a = yre[t], q = yim[t];
    out[((size_t)b * 1025 + f) * 16 + t] = sqrtf(a * a + q * q);
  }
}

// ---------------------------------------------------------------------------
extern "C" void kernel_launch(void* const* d_in, const int* in_sizes, int n_in,
                              void* d_out, int out_size, void* d_ws, size_t ws_size,
                              hipStream_t stream)
{
  const float* data0 = (const float*)d_in[0];
  const float* z     = (const float*)d_in[1];
  const float* w[4]  = {(const float*)d_in[2], (const float*)d_in[6],
                        (const float*)d_in[10], (const float*)d_in[14]};
  const float* bs[4] = {(const float*)d_in[3], (const float*)d_in[7],
                        (const float*)d_in[11], (const float*)d_in[15]};
  const float* gs[4] = {(const float*)d_in[4], (const float*)d_in[8],
                        (const float*)d_in[12], (const float*)d_in[16]};
  const float* es[4] = {(const float*)d_in[5], (const float*)d_in[9],
                        (const float*)d_in[13], (const float*)d_in[17]};
  const float* wf    = (const float*)d_in[18];
  const float* bfv   = (const float*)d_in[19];
  const int* delays  = (const int*)d_in[20];
  float* out = (float*)d_out;

  // workspace layout (floats): 2 activation ping-pong bufs + coefs + head + DFT
  float* bufA = (float*)d_ws;               // 16384*256
  float* bufB = bufA + 4194304;             // 16384*256
  float* coef = bufB + 4194304;             // 64*2*256*6 = 196608
  float* pbuf = coef + 196608;              // 16384*4
  float* Xre  = pbuf + 65536;
  float* Xim  = Xre + 16400;
  float* rre  = Xim + 16400;
  float* rim  = rre + 16400;
  float* dre  = rim + 16400;
  float* dimv = dre + 16400;

  dim3 blk(256);
  gemm_wmma_f32<1><<<dim3(4, 256), blk, 0, stream>>>(data0, data0, w[0], bs[0], bufA, 256, 1025);
  ln_relu<256><<<2048, blk, 0, stream>>>(bufA, gs[0], es[0]);
  gemm_wmma_f32<0><<<dim3(4, 256), blk, 0, stream>>>(bufA, nullptr, w[1], bs[1], bufB, 256, 256);
  ln_relu<256><<<2048, blk, 0, stream>>>(bufB, gs[1], es[1]);
  gemm_wmma_f32<0><<<dim3(2, 256), blk, 0, stream>>>(bufB, nullptr, w[2], bs[2], bufA, 128, 256);
  ln_relu<128><<<2048, blk, 0, stream>>>(bufA, gs[2], es[2]);
  gemm_wmma_f32<0><<<dim3(1, 256), blk, 0, stream>>>(bufA, nullptr, w[3], bs[3], bufB, 64, 128);
  ln_relu<64><<<2048, blk, 0, stream>>>(bufB, gs[3], es[3]);
  head_gemv<<<64, blk, 0, stream>>>(bufB, wf, bfv, pbuf);
  biquad_coef<<<128, blk, 0, stream>>>(pbuf, coef);
  rfft_rhs<<<1025, blk, 0, stream>>>(z, delays, Xre, Xim, rre, rim, dre, dimv);
  biquad_solve<<<dim3(1025, 64), blk, 0, stream>>>(coef, rre, rim, dre, dimv, out);
}